// PNAConv_88364657148501
// MI455X (gfx1250) — compile-verified
//
#include <hip/hip_runtime.h>
#include <math.h>

#define DIM 128
#define NNODES 50000
#define NEDGES 800000

typedef __attribute__((ext_vector_type(2))) float v2f;
typedef __attribute__((ext_vector_type(8))) float v8f;

__device__ __forceinline__ v8f wmma_f32(v2f a, v2f b, v8f c) {
  // V_WMMA_F32_16X16X4_F32: D = A(16x4) * B(4x16) + C(16x16), fp32 throughout
  return __builtin_amdgcn_wmma_f32_16x16x4_f32(
      /*neg_a=*/false, a, /*neg_b=*/false, b,
      /*c_mod=*/(short)0, c, /*reuse_a=*/false, /*reuse_b=*/false);
}

// ---------------------------------------------------------------------------
// Kernel 0: (re)initialize scatter accumulators every launch (graph-replayed).
// ---------------------------------------------------------------------------
__global__ __launch_bounds__(256) void init_ws(float* __restrict__ S,
                                               unsigned int* __restrict__ Mx,
                                               int* __restrict__ cnt) {
  int i = blockIdx.x * 256 + threadIdx.x;
  if (i < NNODES * DIM) {
    S[i] = 0.0f;
    Mx[i] = 0xFF7FFFFFu;  // bits of -FLT_MAX
  }
  if (i < NNODES) cnt[i] = 0;
}

// ---------------------------------------------------------------------------
// Kernel 1: node-level pre-projections.
//   P = x @ pre_W[0:128, :]          (multiplies the x[dst] half of concat)
//   Q = x @ pre_W[128:256, :] + b    (multiplies the x[src] half; bias folded)
// 8 wave32s/block; each wave computes one 16x16 column tile of P and Q.
// ---------------------------------------------------------------------------
__global__ __launch_bounds__(256) void node_pre_gemm(
    const float* __restrict__ x, const float* __restrict__ preW,
    const float* __restrict__ preb, float* __restrict__ P,
    float* __restrict__ Q) {
  __shared__ float xs[16][DIM + 4];
  const int tid = threadIdx.x;
  const int row0 = blockIdx.x * 16;

  for (int i = tid; i < 16 * DIM; i += 256) {
    int r = i >> 7, c = i & (DIM - 1);
    xs[r][c] = x[(row0 + r) * DIM + c];
  }
  __syncthreads();

  const int lane = tid & 31;
  const int wid = tid >> 5;    // 0..7 -> output column tile
  const int half = lane >> 4;  // 0/1
  const int mrow = lane & 15;
  const int n0 = wid * 16;
  const int col = n0 + mrow;

  v8f accP = {};
  v8f accQ = {};
  for (int k0 = 0; k0 < DIM; k0 += 4) {
    const int kA = k0 + 2 * half;
    v2f a;
    a.x = xs[mrow][kA];
    a.y = xs[mrow][kA + 1];
    v2f bP, bQ;
    bP.x = preW[(kA + 0) * DIM + col];
    bP.y = preW[(kA + 1) * DIM + col];
    bQ.x = preW[(DIM + kA + 0) * DIM + col];
    bQ.y = preW[(DIM + kA + 1) * DIM + col];
    accP = wmma_f32(a, bP, accP);  // same A fragment reused for both GEMMs
    accQ = wmma_f32(a, bQ, accQ);
  }

  const float bq = preb[col];
#pragma unroll
  for (int i = 0; i < 8; ++i) {
    const int r = row0 + i + 8 * half;
    P[r * DIM + col] = accP[i];
    Q[r * DIM + col] = accQ[i] + bq;
  }
}

// ---------------------------------------------------------------------------
// Kernel 2: per-edge message + multi-aggregator scatter.
//   m_e = P[dst] + Q[src];  atomic sum, atomic float-max, atomic count.
// 128 threads per edge (one per feature), fully coalesced; P/Q/S/Mx are
// 25.6 MB each -> resident in the 192 MB L2, atomics resolve in L2.
// ---------------------------------------------------------------------------
__global__ __launch_bounds__(256) void edge_scatter(
    const int* __restrict__ ei, const float* __restrict__ P,
    const float* __restrict__ Q, float* __restrict__ S,
    unsigned int* __restrict__ Mx, int* __restrict__ cnt) {
  const unsigned int gid = blockIdx.x * 256u + threadIdx.x;
  const int e = (int)(gid >> 7);
  const int f = (int)(gid & 127u);
  if (e >= NEDGES) return;

  const int s = ei[e];           // edge_index row 0 = src
  const int d = ei[NEDGES + e];  // edge_index row 1 = dst

  const float m = P[d * DIM + f] + Q[s * DIM + f];

  atomicAdd(&S[d * DIM + f], m);
  // sign-aware float max: positives via signed int max, negatives via uint min
  if (m >= 0.0f)
    atomicMax((int*)&Mx[d * DIM + f], __float_as_int(m));
  else
    atomicMin(&Mx[d * DIM + f], __float_as_uint(m));
  if (f == 0) atomicAdd(&cnt[d], 1);
}

// ---------------------------------------------------------------------------
// Kernel 3: node update.
//   u   = concat(x, mean, max, sum) @ post_W + post_b      (K = 512)
//   out = relu(u @ lin_W + lin_b)                          (K = 128)
// concat tile built in LDS (mean/max derived from S, cnt, Mx), u staged in
// LDS between the two WMMA GEMMs.
// ---------------------------------------------------------------------------
__global__ __launch_bounds__(256) void node_post(
    const float* __restrict__ x, const float* __restrict__ S,
    const unsigned int* __restrict__ Mx, const int* __restrict__ cnt,
    const float* __restrict__ postW, const float* __restrict__ postb,
    const float* __restrict__ linW, const float* __restrict__ linb,
    float* __restrict__ out) {
  __shared__ float as[16][512 + 4];
  __shared__ float us[16][DIM + 4];
  const int tid = threadIdx.x;
  const int row0 = blockIdx.x * 16;

  for (int i = tid; i < 16 * 512; i += 256) {
    const int r = i >> 9, c = i & 511;
    const int sec = c >> 7, f = c & 127;
    const int node = row0 + r;
    float v;
    if (sec == 0) {
      v = x[node * DIM + f];
    } else {
      const float sv = S[node * DIM + f];
      const int cn = cnt[node];
      if (sec == 1)
        v = sv / (float)max(cn, 1);                              // mean
      else if (sec == 2)
        v = (cn > 0) ? __uint_as_float(Mx[node * DIM + f]) : 0.0f;  // max
      else
        v = sv;                                                  // sum
    }
    as[r][c] = v;
  }
  __syncthreads();

  const int lane = tid & 31;
  const int wid = tid >> 5;
  const int half = lane >> 4;
  const int mrow = lane & 15;
  const int n0 = wid * 16;
  const int col = n0 + mrow;

  // u tile: [16 x 512] @ post_W[512 x 128], this wave's 16 columns
  v8f acc = {};
  for (int k0 = 0; k0 < 512; k0 += 4) {
    const int kA = k0 + 2 * half;
    v2f a;
    a.x = as[mrow][kA];
    a.y = as[mrow][kA + 1];
    v2f b;
    b.x = postW[(kA + 0) * DIM + col];
    b.y = postW[(kA + 1) * DIM + col];
    acc = wmma_f32(a, b, acc);
  }
  const float pb = postb[col];
#pragma unroll
  for (int i = 0; i < 8; ++i) us[i + 8 * half][col] = acc[i] + pb;
  __syncthreads();

  // out tile: [16 x 128] @ lin_W[128 x 128] + bias, relu
  v8f acc2 = {};
  for (int k0 = 0; k0 < DIM; k0 += 4) {
    const int kA = k0 + 2 * half;
    v2f a;
    a.x = us[mrow][kA];
    a.y = us[mrow][kA + 1];
    v2f b;
    b.x = linW[(kA + 0) * DIM + col];
    b.y = linW[(kA + 1) * DIM + col];
    acc2 = wmma_f32(a, b, acc2);
  }
  const float lb = linb[col];
#pragma unroll
  for (int i = 0; i < 8; ++i) {
    const float v = acc2[i] + lb;
    out[(row0 + i + 8 * half) * DIM + col] = fmaxf(v, 0.0f);
  }
}

// ---------------------------------------------------------------------------
extern "C" void kernel_launch(void* const* d_in, const int* in_sizes, int n_in,
                              void* d_out, int out_size, void* d_ws,
                              size_t ws_size, hipStream_t stream) {
  const float* x = (const float*)d_in[0];
  const int* ei = (const int*)d_in[1];
  const float* preW = (const float*)d_in[2];
  const float* preb = (const float*)d_in[3];
  const float* postW = (const float*)d_in[4];
  const float* postb = (const float*)d_in[5];
  const float* linW = (const float*)d_in[6];
  const float* linb = (const float*)d_in[7];
  float* out = (float*)d_out;

  // workspace layout: P | Q | S | Mx | cnt   (~102.6 MB)
  float* P = (float*)d_ws;
  float* Q = P + (size_t)NNODES * DIM;
  float* S = Q + (size_t)NNODES * DIM;
  unsigned int* Mx = (unsigned int*)(S + (size_t)NNODES * DIM);
  int* cnt = (int*)(Mx + (size_t)NNODES * DIM);

  const int initBlocks = (NNODES * DIM + 255) / 256;  // 25000
  init_ws<<<initBlocks, 256, 0, stream>>>(S, Mx, cnt);

  node_pre_gemm<<<NNODES / 16, 256, 0, stream>>>(x, preW, preb, P, Q);

  const int edgeBlocks = (NEDGES * (DIM / 2) + 255) / 256 * 2 / 2;  // E*64/256 (unused)
  edge_scatter<<<(NEDGES / 2), 256, 0, stream>>>(ei, P, Q, S, Mx, cnt);
  (void)edgeBlocks;

  node_post<<<NNODES / 16, 256, 0, stream>>>(x, S, Mx, cnt, postW, postb, linW,
                                             linb, out);
}